// PEGCN_53824530153898
// MI455X (gfx1250) — compile-verified
//
#include <hip/hip_runtime.h>

// ---------------------------------------------------------------------------
// PE-GCN forward on MI455X (gfx1250, wave32).
//   spenc MLP (2->128->16) + GCNConv(19->32) + GCNConv(32->32) + fc(32->1)
// Dense GEMMs use V_WMMA_F32_16X16X4_F32 (full f32 precision -- GEMM flops are
// negligible vs. the L2-resident edge scatter, so no reason to quantize).
// Edge scatter: one wave per edge, lane == feature channel (H=32 == wave32),
// coalesced 128B reads + global_atomic_add_f32 bursts, all L2-resident.
// ---------------------------------------------------------------------------

typedef __attribute__((ext_vector_type(2))) float v2f;
typedef __attribute__((ext_vector_type(8))) float v8f;

#define FIN   3
#define EHID  128
#define EDIM  16
#define FPAD  20    // 3 + 16 = 19, padded to multiple of 4
#define HW    32    // GCN hidden width

__device__ __forceinline__ v8f wmma_f32_16x16x4(v2f a, v2f b, v8f c) {
  // 8 args: (neg_a, A, neg_b, B, c_mod, C, reuse_a, reuse_b)
  return __builtin_amdgcn_wmma_f32_16x16x4_f32(false, a, false, b,
                                               (short)0, c, false, false);
}

// ---------------- spenc: emb = relu(c@We1+be1)@We2+be2 ; feat=[x|emb|0] -----
__global__ __launch_bounds__(32)
void spenc_kernel(const float* __restrict__ x, const float* __restrict__ c,
                  const float* __restrict__ We1, const float* __restrict__ be1,
                  const float* __restrict__ We2, const float* __restrict__ be2,
                  float* __restrict__ feat) {
  __shared__ float hid[16 * EHID];          // 8 KB: relu hidden for 16 nodes
  const int lane = threadIdx.x;
  const int m0   = blockIdx.x * 16;

  // hidden layer: hid[m][j] = relu(c0*We1[0][j] + c1*We1[1][j] + be1[j])
  for (int idx = lane; idx < 16 * EHID; idx += 32) {
    int m = idx >> 7, j = idx & (EHID - 1);
    float c0 = c[(m0 + m) * 2 + 0];
    float c1 = c[(m0 + m) * 2 + 1];
    float h  = fmaf(c0, We1[j], fmaf(c1, We1[EHID + j], be1[j]));
    hid[m * EHID + j] = h > 0.f ? h : 0.f;
  }
  __syncthreads();

  // 16x16 emb tile = hid[16x128] @ We2[128x16], K-chunks of 4 via WMMA f32.
  const int n     = lane & 15;              // N (and A-row M) index
  const int khalf = (lane >> 4) << 1;       // lanes 16-31 handle K+2,K+3
  v8f acc;
  const float bias = be2[n];
#pragma unroll
  for (int i = 0; i < 8; ++i) acc[i] = bias;   // C seeded with be2 broadcast

  for (int k0 = 0; k0 < EHID; k0 += 4) {
    const int kb = k0 + khalf;
    v2f a, b;
    a.x = hid[n * EHID + kb];
    a.y = hid[n * EHID + kb + 1];
    b.x = We2[kb * EDIM + n];
    b.y = We2[(kb + 1) * EDIM + n];
    acc = wmma_f32_16x16x4(a, b, acc);
  }

  // store emb into feat[:, 3..18]  (C/D layout: VGPR i -> M = i + 8*(lane>=16))
  const int mo = (lane >> 4) * 8;
#pragma unroll
  for (int i = 0; i < 8; ++i)
    feat[(m0 + mo + i) * FPAD + FIN + n] = acc[i];

  // store raw features + zero the pad column
  for (int idx = lane; idx < 16 * 4; idx += 32) {
    int mm = idx >> 2, s = idx & 3;
    feat[(m0 + mm) * FPAD + (s < 3 ? s : FPAD - 1)] =
        (s < 3) ? x[(m0 + mm) * FIN + s] : 0.f;
  }
}

// ---------------- generic WMMA tile GEMM: C[Mx Nout] = A[MxK] @ B[KxNout] ---
__global__ __launch_bounds__(32)
void gemm_wmma_kernel(const float* __restrict__ A, const float* __restrict__ B,
                      float* __restrict__ C, int K, int Nout) {
  __shared__ float As[16 * 32];             // K <= 32, padded stride 32
  const int lane = threadIdx.x;
  const int m0   = blockIdx.x * 16;
  const int n0   = blockIdx.y * 16;

  for (int idx = lane; idx < 16 * K; idx += 32) {
    int m = idx / K, k = idx - m * K;
    As[m * 32 + k] = A[(m0 + m) * K + k];
  }
  __syncthreads();

  const int n     = lane & 15;
  const int khalf = (lane >> 4) << 1;
  v8f acc;
#pragma unroll
  for (int i = 0; i < 8; ++i) acc[i] = 0.f;

  for (int k0 = 0; k0 < K; k0 += 4) {
    const int kb = k0 + khalf;
    v2f a, b;
    a.x = As[n * 32 + kb];
    a.y = As[n * 32 + kb + 1];
    b.x = B[kb * Nout + n0 + n];
    b.y = B[(kb + 1) * Nout + n0 + n];
    acc = wmma_f32_16x16x4(a, b, acc);
  }

  const int mo = (lane >> 4) * 8;
#pragma unroll
  for (int i = 0; i < 8; ++i)
    C[(m0 + mo + i) * Nout + n0 + n] = acc[i];
}

// ---------------- small helpers --------------------------------------------
__global__ void pad_w1_kernel(const float* __restrict__ W1, float* __restrict__ W1p) {
  int t = blockIdx.x * blockDim.x + threadIdx.x;
  if (t >= FPAD * HW) return;
  int r = t / HW, cc = t - r * HW;
  W1p[t] = (r < FIN + EDIM) ? W1[r * HW + cc] : 0.f;
}

__global__ void deg_init_kernel(float* __restrict__ deg, int n) {
  int t = blockIdx.x * blockDim.x + threadIdx.x;
  if (t < n) deg[t] = 1.0f;                 // self-loop weight
}

__global__ void deg_acc_kernel(const int* __restrict__ ei,
                               const float* __restrict__ ew,
                               float* __restrict__ deg, int E) {
  int e = blockIdx.x * blockDim.x + threadIdx.x;
  if (e >= E) return;
  atomicAdd(&deg[ei[E + e]], ew[e]);        // weighted in-degree at col
}

__global__ void dinv_kernel(const float* __restrict__ deg,
                            float* __restrict__ dinv, int n) {
  int t = blockIdx.x * blockDim.x + threadIdx.x;
  if (t >= n) return;
  float d = deg[t];
  dinv[t] = d > 0.f ? rsqrtf(d) : 0.f;
}

__global__ void zero_kernel(float* __restrict__ p, int n) {
  int t = blockIdx.x * blockDim.x + threadIdx.x;
  if (t < n) p[t] = 0.f;
}

// one wave per edge; lane == output channel (HW == 32 == wave32)
__global__ __launch_bounds__(256)
void scatter_kernel(const int* __restrict__ ei, const float* __restrict__ ew,
                    const float* __restrict__ dinv, const float* __restrict__ xw,
                    float* __restrict__ agg, int E) {
  int e = blockIdx.x * 8 + (threadIdx.x >> 5);
  int lane = threadIdx.x & 31;
  if (e >= E) return;
  int r  = ei[e];
  int cc = ei[E + e];
  float norm = dinv[r] * ew[e] * dinv[cc];
  atomicAdd(&agg[cc * HW + lane], xw[r * HW + lane] * norm);
}

// fuse self-loop contribution (norm = dinv[n]^2 * 1) + bias + relu
__global__ void finalize_kernel(const float* __restrict__ agg,
                                const float* __restrict__ xw,
                                const float* __restrict__ dinv,
                                const float* __restrict__ b,
                                float* __restrict__ h, int n) {
  int t = blockIdx.x * blockDim.x + threadIdx.x;
  if (t >= n * HW) return;
  int nd = t >> 5, j = t & 31;
  float d = dinv[nd];
  float v = agg[t] + d * d * xw[t] + b[j];
  h[t] = v > 0.f ? v : 0.f;
}

__global__ void fc_kernel(const float* __restrict__ h,
                          const float* __restrict__ Wfc,
                          const float* __restrict__ bfc,
                          float* __restrict__ out, int n) {
  int nd = blockIdx.x * blockDim.x + threadIdx.x;
  if (nd >= n) return;
  float s = bfc[0];
#pragma unroll
  for (int j = 0; j < HW; ++j) s = fmaf(h[nd * HW + j], Wfc[j], s);
  out[nd] = s;
}

// ---------------------------------------------------------------------------
extern "C" void kernel_launch(void* const* d_in, const int* in_sizes, int n_in,
                              void* d_out, int out_size, void* d_ws, size_t ws_size,
                              hipStream_t stream) {
  const float* x   = (const float*)d_in[0];
  const float* c   = (const float*)d_in[1];
  const int*   ei  = (const int*)  d_in[2];
  const float* ew  = (const float*)d_in[3];
  const float* We1 = (const float*)d_in[4];
  const float* be1 = (const float*)d_in[5];
  const float* We2 = (const float*)d_in[6];
  const float* be2 = (const float*)d_in[7];
  const float* W1  = (const float*)d_in[8];
  const float* b1  = (const float*)d_in[9];
  const float* W2  = (const float*)d_in[10];
  const float* b2  = (const float*)d_in[11];
  const float* Wfc = (const float*)d_in[12];
  const float* bfc = (const float*)d_in[13];
  float* out = (float*)d_out;

  const int N = in_sizes[0] / FIN;          // 100000 (divisible by 16)
  const int E = in_sizes[3];                // 2000000 (divisible by 8)

  // workspace layout (floats)
  float* feat = (float*)d_ws;               // N*FPAD
  float* W1p  = feat + (size_t)N * FPAD;    // FPAD*HW
  float* deg  = W1p + FPAD * HW;            // N
  float* dinv = deg + N;                    // N
  float* xw   = dinv + N;                   // N*HW
  float* agg  = xw + (size_t)N * HW;        // N*HW
  float* h1   = agg + (size_t)N * HW;       // N*HW
  float* h2   = h1 + (size_t)N * HW;        // N*HW

  const int T = 256;
  // 1) positional encoder + feature concat (WMMA 16x16x4 f32)
  spenc_kernel<<<N / 16, 32, 0, stream>>>(x, c, We1, be1, We2, be2, feat);
  // 2) pad W1 to [20][32]
  pad_w1_kernel<<<(FPAD * HW + T - 1) / T, T, 0, stream>>>(W1, W1p);
  // 3) normalization: weighted degree (w/ self-loop) -> rsqrt
  deg_init_kernel<<<(N + T - 1) / T, T, 0, stream>>>(deg, N);
  deg_acc_kernel<<<(E + T - 1) / T, T, 0, stream>>>(ei, ew, deg, E);
  dinv_kernel<<<(N + T - 1) / T, T, 0, stream>>>(deg, dinv, N);

  // ---- GCN layer 1: feat[N][20] @ W1p[20][32] -> scatter -> relu ----
  gemm_wmma_kernel<<<dim3(N / 16, HW / 16), 32, 0, stream>>>(feat, W1p, xw, FPAD, HW);
  zero_kernel<<<(N * HW + T - 1) / T, T, 0, stream>>>(agg, N * HW);
  scatter_kernel<<<E / 8, T, 0, stream>>>(ei, ew, dinv, xw, agg, E);
  finalize_kernel<<<(N * HW + T - 1) / T, T, 0, stream>>>(agg, xw, dinv, b1, h1, N);

  // ---- GCN layer 2: h1[N][32] @ W2[32][32] -> scatter -> relu ----
  gemm_wmma_kernel<<<dim3(N / 16, HW / 16), 32, 0, stream>>>(h1, W2, xw, HW, HW);
  zero_kernel<<<(N * HW + T - 1) / T, T, 0, stream>>>(agg, N * HW);
  scatter_kernel<<<E / 8, T, 0, stream>>>(ei, ew, dinv, xw, agg, E);
  finalize_kernel<<<(N * HW + T - 1) / T, T, 0, stream>>>(agg, xw, dinv, b2, h2, N);

  // ---- head: h2 @ Wfc + bfc ----
  fc_kernel<<<(N + T - 1) / T, T, 0, stream>>>(h2, Wfc, bfc, out, N);
}